// MyGNN_2379411882474
// MI455X (gfx1250) — compile-verified
//
#include <hip/hip_runtime.h>

#define N_NODES 50000
#define N_EDGES 800000
#define HID 64

typedef float v2f __attribute__((ext_vector_type(2)));
typedef float v8f __attribute__((ext_vector_type(8)));

__device__ __forceinline__ void atomAddF(float* p, float v) {
  // hardware global_atomic_add_f32 (round-to-nearest-even in L2, matches f32 ref closely)
  unsafeAtomicAdd(p, v);
}

// ---------------- degree / norm precompute ----------------

__global__ void k_deg_init(float* __restrict__ deg, int n) {
  int i = blockIdx.x * blockDim.x + threadIdx.x;
  if (i < n) deg[i] = 1.0f;  // self-loop
}

__global__ void k_deg_edges(const int* __restrict__ dst, float* __restrict__ deg, int e) {
  int i = blockIdx.x * blockDim.x + threadIdx.x;
  if (i < e) atomAddF(&deg[dst[i]], 1.0f);
}

__global__ void k_dinv(float* __restrict__ deg, int n) {
  int i = blockIdx.x * blockDim.x + threadIdx.x;
  if (i < n) deg[i] = rsqrtf(deg[i]);  // deg >= 1
}

__global__ void k_norm(const int* __restrict__ src, const int* __restrict__ dst,
                       const float* __restrict__ dinv, float* __restrict__ norm, int e) {
  int i = blockIdx.x * blockDim.x + threadIdx.x;
  if (i < e) norm[i] = dinv[src[i]] * dinv[dst[i]];
}

// ---------------- WMMA f32 GEMM: Lin[M,64] = act(Hin)[M,K] @ W[K,64] ----------------
// 256 threads = 8 waves; waves arranged 2 (M) x 4 (N) 16x16 tiles per block.
// V_WMMA_F32_16X16X4_F32, f32 accumulate — matches the f32 reference exactly in precision class.

template <int K, bool RELU_IN>
__global__ __launch_bounds__(256) void k_gemm(const float* __restrict__ Hin,
                                              const float* __restrict__ W,
                                              float* __restrict__ Lin, int M) {
  const int lane = threadIdx.x & 31;
  const int wave = threadIdx.x >> 5;         // 0..7
  const int mt   = blockIdx.x * 2 + (wave >> 2);
  const int nt   = wave & 3;                 // 0..3 -> col tile
  const int row0 = mt * 16;
  if (row0 >= M) return;                     // wave-uniform: EXEC stays all-ones per wave

  const int m    = lane & 15;                // A row within tile
  const int kb   = (lane >> 4) << 1;         // K sub-offset {0,2}
  const int col0 = nt * 16;
  const int n    = lane & 15;                // B col within tile

  v8f acc = {};
#pragma unroll
  for (int k0 = 0; k0 < K; k0 += 4) {
    const float* ap = Hin + (size_t)(row0 + m) * K + (k0 + kb);
    v2f a;
    a.x = ap[0];
    a.y = ap[1];
    if (RELU_IN) { a.x = fmaxf(a.x, 0.0f); a.y = fmaxf(a.y, 0.0f); }
    const float* bp = W + (size_t)(k0 + kb) * HID + col0 + n;
    v2f b;
    b.x = bp[0];
    b.y = bp[HID];
    acc = __builtin_amdgcn_wmma_f32_16x16x4_f32(
        /*neg_a=*/false, a, /*neg_b=*/false, b,
        /*c_mod=*/(short)0, acc, /*reuse_a=*/false, /*reuse_b=*/false);
  }

  // C/D layout: VGPR j -> row = row0 + j + 8*(lane>>4), col = col0 + (lane&15)
  const int rbase = row0 + ((lane >> 4) << 3);
  const int c     = col0 + (lane & 15);
#pragma unroll
  for (int j = 0; j < 8; ++j) {
    Lin[(size_t)(rbase + j) * HID + c] = acc[j];
  }
}

// ---------------- aggregation: agg = D^-1/2 (A+I) D^-1/2 Lin + b ----------------

__global__ void k_agg_init(const float* __restrict__ lin, const float* __restrict__ dinv,
                           const float* __restrict__ bias, float* __restrict__ agg, int total) {
  int idx = blockIdx.x * blockDim.x + threadIdx.x;
  if (idx >= total) return;
  int i = idx >> 6;
  int f = idx & (HID - 1);
  float di = dinv[i];
  agg[idx] = di * di * lin[idx] + bias[f];   // self-loop term + bias
}

__global__ void k_agg_edges(const int* __restrict__ src, const int* __restrict__ dst,
                            const float* __restrict__ norm, const float* __restrict__ lin,
                            float* __restrict__ agg, int total) {
  int idx = blockIdx.x * blockDim.x + threadIdx.x;
  if (idx >= total) return;
  int e = idx >> 6;                          // 64 consecutive lanes share an edge:
  int f = idx & (HID - 1);                   // broadcast src/dst/norm, coalesced features
  int s = src[e], d = dst[e];
  atomAddF(&agg[(size_t)d * HID + f], norm[e] * lin[(size_t)s * HID + f]);
}

// ---------------- output head: s[i] = relu(h[i]) . W_out ----------------

__global__ void k_out_dot(const float* __restrict__ h, const float* __restrict__ Wout,
                          float* __restrict__ s, int n) {
  int i = blockIdx.x * blockDim.x + threadIdx.x;
  if (i >= n) return;
  const float4* hp = (const float4*)(h + (size_t)i * HID);
  const float4* wp = (const float4*)Wout;
  float acc = 0.0f;
#pragma unroll
  for (int k = 0; k < HID / 4; ++k) {
    float4 v = hp[k];
    float4 w = wp[k];
    acc += fmaxf(v.x, 0.0f) * w.x + fmaxf(v.y, 0.0f) * w.y +
           fmaxf(v.z, 0.0f) * w.z + fmaxf(v.w, 0.0f) * w.w;
  }
  s[i] = acc;
}

__global__ void k_out_init(const float* __restrict__ s, const float* __restrict__ dinv,
                           const float* __restrict__ bout, float* __restrict__ out, int n) {
  int i = blockIdx.x * blockDim.x + threadIdx.x;
  if (i >= n) return;
  float di = dinv[i];
  out[i] = di * di * s[i] + bout[0];
}

__global__ void k_out_edges(const int* __restrict__ src, const int* __restrict__ dst,
                            const float* __restrict__ norm, const float* __restrict__ s,
                            float* __restrict__ out, int e) {
  int i = blockIdx.x * blockDim.x + threadIdx.x;
  if (i >= e) return;
  atomAddF(&out[dst[i]], norm[i] * s[src[i]]);
}

// ---------------- driver ----------------

static inline int cdiv(int a, int b) { return (a + b - 1) / b; }

extern "C" void kernel_launch(void* const* d_in, const int* in_sizes, int n_in,
                              void* d_out, int out_size, void* d_ws, size_t ws_size,
                              hipStream_t stream) {
  (void)in_sizes; (void)n_in; (void)out_size; (void)ws_size;

  const float* x     = (const float*)d_in[0];            // [N,4]
  const int*   ei    = (const int*)d_in[1];              // [2,E]
  const float* W_in  = (const float*)d_in[2];            // [4,64]
  const float* b_in  = (const float*)d_in[3];            // [64]
  const float* W_h   = (const float*)d_in[4];            // [3,64,64]
  const float* b_h   = (const float*)d_in[5];            // [3,64]
  const float* W_out = (const float*)d_in[6];            // [64,1]
  const float* b_out = (const float*)d_in[7];            // [1]
  float*       out   = (float*)d_out;                    // [N]

  const int* src = ei;
  const int* dst = ei + N_EDGES;

  float* ws   = (float*)d_ws;
  float* dinv = ws;                                      // N
  float* norm = dinv + N_NODES;                          // E
  float* svec = norm + N_EDGES;                          // N
  float* bufA = svec + N_NODES;                          // N*64 (lin)
  float* bufB = bufA + (size_t)N_NODES * HID;            // N*64 (agg ping)
  float* bufC = bufB + (size_t)N_NODES * HID;            // N*64 (agg pong)

  const int BT = 256;
  const int nodeBlk = cdiv(N_NODES, BT);
  const int edgeBlk = cdiv(N_EDGES, BT);
  const int featTot = N_NODES * HID;
  const int featBlk = cdiv(featTot, BT);
  const int edgeFeatTot = N_EDGES * HID;                 // 51.2M, fits int
  const int edgeFeatBlk = cdiv(edgeFeatTot, BT);
  const int gemmBlk = cdiv(N_NODES, 32);                 // 2 M-tiles per block

  // ---- GCN norm ----
  k_deg_init<<<nodeBlk, BT, 0, stream>>>(dinv, N_NODES);
  k_deg_edges<<<edgeBlk, BT, 0, stream>>>(dst, dinv, N_EDGES);
  k_dinv<<<nodeBlk, BT, 0, stream>>>(dinv, N_NODES);
  k_norm<<<edgeBlk, BT, 0, stream>>>(src, dst, dinv, norm, N_EDGES);

  // ---- layer 0: x[N,4] @ W_in -> agg(bufB) ----
  k_gemm<4, false><<<gemmBlk, BT, 0, stream>>>(x, W_in, bufA, N_NODES);
  k_agg_init<<<featBlk, BT, 0, stream>>>(bufA, dinv, b_in, bufB, featTot);
  k_agg_edges<<<edgeFeatBlk, BT, 0, stream>>>(src, dst, norm, bufA, bufB, edgeFeatTot);

  // ---- hidden layers 1..3 (ReLU fused into GEMM input load) ----
  float* hin = bufB;
  float* agg = bufC;
  for (int l = 0; l < 3; ++l) {
    k_gemm<HID, true><<<gemmBlk, BT, 0, stream>>>(hin, W_h + (size_t)l * HID * HID, bufA, N_NODES);
    k_agg_init<<<featBlk, BT, 0, stream>>>(bufA, dinv, b_h + (size_t)l * HID, agg, featTot);
    k_agg_edges<<<edgeFeatBlk, BT, 0, stream>>>(src, dst, norm, bufA, agg, edgeFeatTot);
    float* t = hin; hin = agg; agg = t;
  }

  // ---- output head: s = relu(h) . W_out; out = scatter(norm*s) + self-loop + b_out ----
  k_out_dot<<<nodeBlk, BT, 0, stream>>>(hin, W_out, svec, N_NODES);
  k_out_init<<<nodeBlk, BT, 0, stream>>>(svec, dinv, b_out, out, N_NODES);
  k_out_edges<<<edgeBlk, BT, 0, stream>>>(src, dst, norm, svec, out, N_EDGES);
}